// CEAlignment_40570261078040
// MI455X (gfx1250) — compile-verified
//
#include <hip/hip_runtime.h>
#include <hip/hip_bf16.h>

// ---------------------------------------------------------------------------
// CEAlignment on MI455X (gfx1250, wave32, WMMA)
//   B=1024, X1D=X2D=256, HID=512, E=64, C=16, N_SINK=10, EPS=1e-8
// Pipeline:
//   gemm_bias_act (f32 WMMA 16x16x4) x8  -> q1,q2 (1024x1024)
//   norm_rows64                          -> per-(b,c) LayerNorm (ddof=1)
//   affinity (f32 WMMA, batched over c)  -> aff into d_out (b,d,c) + per-c max
//   exp_sub_max                          -> A0 = exp(aff-max) in d_out
//   sinkhorn as u/v scaling-vector iterations (read-only passes over A0)
//   final scale: P = u * A0 * v in place in d_out
// ---------------------------------------------------------------------------

typedef float v2f __attribute__((ext_vector_type(2)));
typedef float v8f __attribute__((ext_vector_type(8)));

#define BSZ   1024
#define CCH   16
#define EDIM  64
#define SLICE (CCH * EDIM)        // 1024 floats per b row of q
#define ATOT  (BSZ * BSZ * CCH)   // 16,777,216 elements of A / P

// ---- order-preserving float<->uint encoding for atomicMax on floats --------
__device__ __forceinline__ unsigned int enc_f32(float f) {
  unsigned int u = __float_as_uint(f);
  return (u & 0x80000000u) ? ~u : (u | 0x80000000u);
}
__device__ __forceinline__ float dec_f32(unsigned int u) {
  return (u & 0x80000000u) ? __uint_as_float(u & 0x7FFFFFFFu)
                           : __uint_as_float(~u);
}

// ---------------------------------------------------------------------------
// GEMM: out[M,N] = act(A[M,K] @ W[K,N] + bias[N]), fp32 via V_WMMA_F32_16X16X4_F32
// Block: 256 threads = 8 waves, tile 128(M) x 64(N); waves 4x2, 32x32 each.
// ---------------------------------------------------------------------------
#define LDA_P 20   // LDS stride for A tile (16 K cols + pad, float4-aligned)
#define LDB_P 68   // LDS stride for B tile (64 N cols + pad, float4-aligned)

__global__ __launch_bounds__(256) void gemm_bias_act(
    const float* __restrict__ A, const float* __restrict__ W,
    const float* __restrict__ bias, float* __restrict__ out,
    int M, int K, int N, int relu)
{
  __shared__ float lA[128 * LDA_P];
  __shared__ float lB[16 * LDB_P];

  const int t    = threadIdx.x;
  const int lane = t & 31;
  const int wave = t >> 5;
  const int wm   = wave & 3;       // 0..3 -> M offset 32*wm
  const int wn   = wave >> 2;      // 0..1 -> N offset 32*wn
  const int bm   = blockIdx.y * 128;
  const int bn   = blockIdx.x * 64;

  v8f acc[2][2] = {};

  const int mrow = lane & 15;           // A frag: M row
  const int koff = (lane >> 4) * 2;     // A/B frag: k pair base (0 or 2)
  const int ncol = lane & 15;           // B frag: N col

  for (int k0 = 0; k0 < K; k0 += 16) {
    // stage A tile 128x16 (512 float4, 2 per thread)
    #pragma unroll
    for (int i = 0; i < 2; ++i) {
      int idx = t + i * 256;            // 0..511
      int row = idx >> 2;               // 4 float4 per row
      int c4  = idx & 3;
      float4 v = *(const float4*)(A + (size_t)(bm + row) * K + k0 + c4 * 4);
      *(float4*)(lA + row * LDA_P + c4 * 4) = v;
    }
    // stage B tile 16x64 (256 float4, 1 per thread)
    {
      int row = t >> 4;
      int c4  = t & 15;
      float4 v = *(const float4*)(W + (size_t)(k0 + row) * N + bn + c4 * 4);
      *(float4*)(lB + row * LDB_P + c4 * 4) = v;
    }
    __syncthreads();

    #pragma unroll
    for (int kk = 0; kk < 16; kk += 4) {
      v2f afrag[2], bfrag[2];
      #pragma unroll
      for (int mt = 0; mt < 2; ++mt) {
        int row = wm * 32 + mt * 16 + mrow;
        afrag[mt] = *(const v2f*)(lA + row * LDA_P + kk + koff);
      }
      #pragma unroll
      for (int nt = 0; nt < 2; ++nt) {
        int col = wn * 32 + nt * 16 + ncol;
        bfrag[nt].x = lB[(kk + koff + 0) * LDB_P + col];
        bfrag[nt].y = lB[(kk + koff + 1) * LDB_P + col];
      }
      #pragma unroll
      for (int mt = 0; mt < 2; ++mt)
        #pragma unroll
        for (int nt = 0; nt < 2; ++nt)
          acc[mt][nt] = __builtin_amdgcn_wmma_f32_16x16x4_f32(
              false, afrag[mt], false, bfrag[nt],
              (short)0, acc[mt][nt], false, false);
    }
    __syncthreads();
  }

  // epilogue: bias + optional ReLU
  const int rsel = (lane >> 4) * 8;     // C/D layout: M = v + 8*(lane>>4)
  #pragma unroll
  for (int nt = 0; nt < 2; ++nt) {
    int col = bn + wn * 32 + nt * 16 + ncol;
    float bv = bias[col];
    #pragma unroll
    for (int mt = 0; mt < 2; ++mt) {
      #pragma unroll
      for (int v = 0; v < 8; ++v) {
        int row = bm + wm * 32 + mt * 16 + rsel + v;
        float x = acc[mt][nt][v] + bv;
        if (relu) x = fmaxf(x, 0.0f);
        out[(size_t)row * N + col] = x;
      }
    }
  }
}

// ---------------------------------------------------------------------------
// Per-row normalization of q (16384 rows of 64), mean/var (ddof=1), in place.
// One wave per row; 2 elements per lane; shfl_xor reduction.
// ---------------------------------------------------------------------------
__global__ __launch_bounds__(256) void norm_rows64(float* __restrict__ q)
{
  const int lane = threadIdx.x & 31;
  const int wave = threadIdx.x >> 5;
  const int row  = blockIdx.x * 8 + wave;
  float* p = q + (size_t)row * 64;

  float2 x = *(float2*)(p + lane * 2);
  float s  = x.x + x.y;
  float ss = x.x * x.x + x.y * x.y;
  #pragma unroll
  for (int off = 16; off > 0; off >>= 1) {
    s  += __shfl_xor(s,  off, 32);
    ss += __shfl_xor(ss, off, 32);
  }
  float mean = s * (1.0f / 64.0f);
  float var  = (ss - 64.0f * mean * mean) * (1.0f / 63.0f);
  float inv  = rsqrtf(var + 1e-8f);
  x.x = (x.x - mean) * inv;
  x.y = (x.y - mean) * inv;
  *(float2*)(p + lane * 2) = x;
}

// ---------------------------------------------------------------------------
// Init: v = 1, per-channel max accumulators = 0 (encoded lower bound)
// ---------------------------------------------------------------------------
__global__ void sink_init(float* __restrict__ v, unsigned int* __restrict__ maxbuf)
{
  int i = blockIdx.x * 256 + threadIdx.x;
  if (i < BSZ * CCH) v[i] = 1.0f;
  if (i < CCH) maxbuf[i] = 0u;
}

// ---------------------------------------------------------------------------
// Affinity: aff[b,d,c] = (1/8) * sum_e q1n[b,c,e]*q2n[d,c,e], written to
// d_out in (b,d,c) layout; also per-c running max via encoded atomicMax.
// Batched fp32 WMMA GEMM: block tile 128(b) x 64(d), one c per block.z.
// ---------------------------------------------------------------------------
#define LQA_P 68   // A tile stride (64 e cols + pad)
#define LQB_P 65   // B tile stride (64 d cols + pad)

__global__ __launch_bounds__(256) void affinity_wmma(
    const float* __restrict__ q1, const float* __restrict__ q2,
    float* __restrict__ aff, unsigned int* __restrict__ maxbuf)
{
  __shared__ float lA[128 * LQA_P];   // 128 b-rows x 64 e
  __shared__ float lB[64 * LQB_P];    // 64 e-rows x 64 d (transposed in)
  __shared__ float red[256];

  const int t    = threadIdx.x;
  const int lane = t & 31;
  const int wave = t >> 5;
  const int wm   = wave & 3;
  const int wn   = wave >> 2;
  const int bb   = blockIdx.y * 128;  // b base
  const int bd   = blockIdx.x * 64;   // d base
  const int c    = blockIdx.z;

  // stage A: q1[(bb+row)][c*64 + e], 2048 float4, 8 per thread
  #pragma unroll
  for (int i = 0; i < 8; ++i) {
    int idx = t + i * 256;
    int row = idx >> 4;               // 16 float4 per row
    int c4  = idx & 15;
    float4 v = *(const float4*)(q1 + (size_t)(bb + row) * SLICE + c * EDIM + c4 * 4);
    *(float4*)(lA + row * LQA_P + c4 * 4) = v;
  }
  // stage B transposed: lB[e][dl] = q2[(bd+dl)][c*64+e]; 1024 float4, 4/thread
  #pragma unroll
  for (int i = 0; i < 4; ++i) {
    int idx = t + i * 256;
    int dl = idx >> 4;
    int e4 = idx & 15;
    float4 v = *(const float4*)(q2 + (size_t)(bd + dl) * SLICE + c * EDIM + e4 * 4);
    lB[(e4 * 4 + 0) * LQB_P + dl] = v.x;
    lB[(e4 * 4 + 1) * LQB_P + dl] = v.y;
    lB[(e4 * 4 + 2) * LQB_P + dl] = v.z;
    lB[(e4 * 4 + 3) * LQB_P + dl] = v.w;
  }
  __syncthreads();

  v8f acc[2][2] = {};
  const int mrow = lane & 15;
  const int koff = (lane >> 4) * 2;
  const int ncol = lane & 15;

  #pragma unroll
  for (int kk = 0; kk < 64; kk += 4) {
    v2f afrag[2], bfrag[2];
    #pragma unroll
    for (int mt = 0; mt < 2; ++mt) {
      int row = wm * 32 + mt * 16 + mrow;
      afrag[mt] = *(const v2f*)(lA + row * LQA_P + kk + koff);
    }
    #pragma unroll
    for (int nt = 0; nt < 2; ++nt) {
      int col = wn * 32 + nt * 16 + ncol;
      bfrag[nt].x = lB[(kk + koff + 0) * LQB_P + col];
      bfrag[nt].y = lB[(kk + koff + 1) * LQB_P + col];
    }
    #pragma unroll
    for (int mt = 0; mt < 2; ++mt)
      #pragma unroll
      for (int nt = 0; nt < 2; ++nt)
        acc[mt][nt] = __builtin_amdgcn_wmma_f32_16x16x4_f32(
            false, afrag[mt], false, bfrag[nt],
            (short)0, acc[mt][nt], false, false);
  }

  // write aff (b,d,c) + local max
  const float scale = 0.125f;          // 1/sqrt(E)
  const int rsel = (lane >> 4) * 8;
  float lmax = -3.4e38f;
  #pragma unroll
  for (int mt = 0; mt < 2; ++mt) {
    #pragma unroll
    for (int nt = 0; nt < 2; ++nt) {
      int col = bd + wn * 32 + nt * 16 + ncol;
      #pragma unroll
      for (int v = 0; v < 8; ++v) {
        int row = bb + wm * 32 + mt * 16 + rsel + v;
        float x = acc[mt][nt][v] * scale;
        lmax = fmaxf(lmax, x);
        aff[((size_t)row * BSZ + col) * CCH + c] = x;
      }
    }
  }
  red[t] = lmax;
  __syncthreads();
  for (int off = 128; off > 0; off >>= 1) {
    if (t < off) red[t] = fmaxf(red[t], red[t + off]);
    __syncthreads();
  }
  if (t == 0) atomicMax(&maxbuf[c], enc_f32(red[0]));
}

// ---------------------------------------------------------------------------
// A = exp(aff - max[c]), elementwise over (b,d,c) in d_out
// ---------------------------------------------------------------------------
__global__ __launch_bounds__(256) void exp_sub_max(
    float* __restrict__ A, const unsigned int* __restrict__ maxbuf)
{
  size_t base   = ((size_t)blockIdx.x * 256 + threadIdx.x) * 4;
  size_t stride = (size_t)gridDim.x * 256 * 4;
  for (size_t i = base; i < (size_t)ATOT; i += stride) {
    #pragma unroll
    for (int j = 0; j < 4; ++j) {
      size_t idx = i + j;
      float mv = dec_f32(maxbuf[idx & 15]);
      A[idx] = expf(A[idx] - mv);
    }
  }
}

// ---------------------------------------------------------------------------
// Sinkhorn half-step 1: u[b,c] = r[b,c] / sum_d A[b,d,c]*v[d,c]
// One block per b; fully contiguous 64 KiB read of the b-slice.
// ---------------------------------------------------------------------------
__global__ __launch_bounds__(256) void sink_u(
    const float* __restrict__ A, const float* __restrict__ v,
    const float* __restrict__ r, float* __restrict__ u)
{
  __shared__ float red[256];
  const int t = threadIdx.x;
  const int b = blockIdx.x;
  const float* Ab = A + (size_t)b * (BSZ * CCH);

  float s = 0.0f;
  #pragma unroll 4
  for (int i = 0; i < 64; ++i) {
    int j = t + 256 * i;               // j = d*16 + c, c == t & 15 (fixed)
    s += Ab[j] * v[j];
  }
  red[t] = s;
  __syncthreads();
  // fold with offsets that are multiples of 16 -> preserves c congruence
  for (int off = 128; off >= 16; off >>= 1) {
    if (t < off) red[t] += red[t + off];
    __syncthreads();
  }
  if (t < CCH) u[b * CCH + t] = r[b * CCH + t] / red[t];
}

// ---------------------------------------------------------------------------
// Sinkhorn half-step 2: v[d,c] = cmarg[d,c] / sum_b A[b,d,c]*u[b,c]
// One block per d; 16 b-streams of 64 B contiguous reads.
// ---------------------------------------------------------------------------
__global__ __launch_bounds__(256) void sink_v(
    const float* __restrict__ A, const float* __restrict__ u,
    const float* __restrict__ cmarg, float* __restrict__ v)
{
  __shared__ float red[256];
  const int t = threadIdx.x;
  const int d = blockIdx.x;
  const int c = t & 15;
  const int bseg = t >> 4;             // 16 segments of 64 b each

  float s = 0.0f;
  #pragma unroll 4
  for (int k = 0; k < 64; ++k) {
    int b = bseg * 64 + k;
    s += A[(size_t)b * (BSZ * CCH) + d * CCH + c] * u[b * CCH + c];
  }
  red[t] = s;
  __syncthreads();
  for (int off = 128; off >= 16; off >>= 1) {
    if (t < off) red[t] += red[t + off];
    __syncthreads();
  }
  if (t < CCH) v[d * CCH + t] = cmarg[d * CCH + t] / red[t];
}

// ---------------------------------------------------------------------------
// Final: P[b,d,c] = u[b,c] * A[b,d,c] * v[d,c], in place in d_out
// ---------------------------------------------------------------------------
__global__ __launch_bounds__(256) void sink_final(
    float* __restrict__ A, const float* __restrict__ u,
    const float* __restrict__ v)
{
  size_t base   = ((size_t)blockIdx.x * 256 + threadIdx.x) * 4;
  size_t stride = (size_t)gridDim.x * 256 * 4;
  for (size_t i = base; i < (size_t)ATOT; i += stride) {
    #pragma unroll
    for (int j = 0; j < 4; ++j) {
      size_t idx = i + j;
      size_t b  = idx >> 14;           // slice = 16384
      size_t dc = idx & 16383;         // d*16 + c
      A[idx] = A[idx] * u[b * CCH + (idx & 15)] * v[dc];
    }
  }
}

// ---------------------------------------------------------------------------
extern "C" void kernel_launch(void* const* d_in, const int* in_sizes, int n_in,
                              void* d_out, int out_size, void* d_ws, size_t ws_size,
                              hipStream_t stream)
{
  const float* x1   = (const float*)d_in[0];
  const float* x2   = (const float*)d_in[1];
  const float* p1   = (const float*)d_in[2];   // r[b,c]
  const float* p2   = (const float*)d_in[3];   // cmarg[d,c]
  const float* w1[4] = {(const float*)d_in[4],  (const float*)d_in[6],
                        (const float*)d_in[8],  (const float*)d_in[10]};
  const float* b1[4] = {(const float*)d_in[5],  (const float*)d_in[7],
                        (const float*)d_in[9],  (const float*)d_in[11]};
  const float* w2[4] = {(const float*)d_in[12], (const float*)d_in[14],
                        (const float*)d_in[16], (const float*)d_in[18]};
  const float* b2[4] = {(const float*)d_in[13], (const float*)d_in[15],
                        (const float*)d_in[17], (const float*)d_in[19]};

  float* ws = (float*)d_ws;
  float* hA = ws;                       // 1024*512
  float* hB = hA + 1024 * 512;          // 1024*512
  float* q1 = hB + 1024 * 512;          // 1024*1024
  float* q2 = q1 + 1024 * 1024;         // 1024*1024
  float* uv_u = q2 + 1024 * 1024;       // 16384
  float* uv_v = uv_u + 16384;           // 16384
  unsigned int* maxbuf = (unsigned int*)(uv_v + 16384);  // 16

  float* Aout = (float*)d_out;          // (b,d,c) = aff -> exp(A) -> P

  // ---- MLP 1: x1 -> q1 ----
  gemm_bias_act<<<dim3(512/64, 1024/128), 256, 0, stream>>>(x1, w1[0], b1[0], hA, 1024, 256, 512, 1);
  gemm_bias_act<<<dim3(512/64, 1024/128), 256, 0, stream>>>(hA, w1[1], b1[1], hB, 1024, 512, 512, 1);
  gemm_bias_act<<<dim3(512/64, 1024/128), 256, 0, stream>>>(hB, w1[2], b1[2], hA, 1024, 512, 512, 1);
  gemm_bias_act<<<dim3(1024/64, 1024/128), 256, 0, stream>>>(hA, w1[3], b1[3], q1, 1024, 512, 1024, 0);
  // ---- MLP 2: x2 -> q2 ----
  gemm_bias_act<<<dim3(512/64, 1024/128), 256, 0, stream>>>(x2, w2[0], b2[0], hA, 1024, 256, 512, 1);
  gemm_bias_act<<<dim3(512/64, 1024/128), 256, 0, stream>>>(hA, w2[1], b2[1], hB, 1024, 512, 512, 1);
  gemm_bias_act<<<dim3(512/64, 1024/128), 256, 0, stream>>>(hB, w2[2], b2[2], hA, 1024, 512, 512, 1);
  gemm_bias_act<<<dim3(1024/64, 1024/128), 256, 0, stream>>>(hA, w2[3], b2[3], q2, 1024, 512, 1024, 0);

  // ---- per-(b,c) normalization over E=64 ----
  norm_rows64<<<16384 / 8, 256, 0, stream>>>(q1);
  norm_rows64<<<16384 / 8, 256, 0, stream>>>(q2);

  // ---- init sinkhorn state (v=1, max accumulators) ----
  sink_init<<<(1024 * 16 + 255) / 256, 256, 0, stream>>>(uv_v, maxbuf);

  // ---- affinity + per-channel max ----
  affinity_wmma<<<dim3(1024/64, 1024/128, CCH), 256, 0, stream>>>(q1, q2, Aout, maxbuf);

  // ---- A = exp(aff - max[c]) ----
  exp_sub_max<<<4096, 256, 0, stream>>>(Aout, maxbuf);

  // ---- Sinkhorn as scaling-vector iterations (read-only passes over A) ----
  for (int it = 0; it < 10; ++it) {
    sink_u<<<1024, 256, 0, stream>>>(Aout, uv_v, p1, uv_u);
    sink_v<<<1024, 256, 0, stream>>>(Aout, uv_u, p2, uv_v);
  }

  // ---- P = u * A * v ----
  sink_final<<<4096, 256, 0, stream>>>(Aout, uv_u, uv_v);
}